// CorrelationFusion_57312043598532
// MI455X (gfx1250) — compile-verified
//
#include <hip/hip_runtime.h>

typedef __attribute__((ext_vector_type(2))) float v2f;
typedef __attribute__((ext_vector_type(8))) float v8f;

#define NB   8
#define TT   8      // T == T_OUT
#define GG   4
#define CPG  64
#define CC   256
#define HH   56
#define WW   56
#define HWSZ (HH*WW)
#define NI   63     // (T-1)*K*K conv inputs per group
#define NO   64     // T_OUT*T conv outputs per group

// -------- Kernel 1: 9-shift cost volume + grouped 1x1 conv (fp32 WMMA) -----
// One block per (n, g, row y).  512 threads = 16 wave32.
// Phase A: corr[px][k] in LDS (px 0..55 valid, k 0..62 valid, rest zero-pad).
// Phase B: (64px x 64k) @ (64k x 64o) via v_wmma_f32_16x16x4_f32, +bias,
//          store wx pixel-major: wx[((n*G+g)*HW + p)*64 + o].
__global__ __launch_bounds__(512) void corrfuse_k1(
    const float* __restrict__ x, const float* __restrict__ conv_w,
    const float* __restrict__ conv_b, float* __restrict__ wx)
{
  __shared__ float corr_s[64 * 65];   // stride 65: conflict-free both phases
  __shared__ float wT_s[64 * 64];     // wT[k][o] = conv_w[g][o][k], row 63 = 0
  __shared__ float bias_s[64];

  const int bid = blockIdx.x;
  const int y = bid % HH;
  const int g = (bid / HH) % GG;
  const int n = bid / (HH * GG);
  const int tid = threadIdx.x;

  // zero corr (covers k=63 column and px>=56 rows)
  for (int i = tid; i < 64 * 65; i += 512) corr_s[i] = 0.f;
  __syncthreads();

  // ---- Phase A: cost volume, thread = (pixel px, frame-pair tp) ----
  const int px = tid & 63;
  const int tp = tid >> 6;            // 0..7, active 0..6

  float acc[9];
#pragma unroll
  for (int s = 0; s < 9; ++s) acc[s] = 0.f;

  if (px < WW && tp < TT - 1) {
    const int ym = (y > 0) ? y - 1 : 0;
    const int yp = (y < HH - 1) ? y + 1 : HH - 1;
    const int rows[3] = { ym * WW, y * WW, yp * WW };
    const int xm = (px > 0) ? px - 1 : 0;
    const int xq = (px < WW - 1) ? px + 1 : WW - 1;
    const int cols[3] = { xm, px, xq };

    const float* xa = x + ((size_t)(n * TT + tp) * CC + g * CPG) * HWSZ + y * WW + px;
    const float* xb = x + ((size_t)(n * TT + tp + 1) * CC + g * CPG) * HWSZ;

    for (int c = 0; c < CPG; ++c) {
      if (c + 4 < CPG) {
        __builtin_prefetch(xa + (size_t)(c + 4) * HWSZ, 0, 0);   // global_prefetch_b8
      }
      const float a = xa[(size_t)c * HWSZ];
      const float* xbc = xb + (size_t)c * HWSZ;
#pragma unroll
      for (int dy = 0; dy < 3; ++dy) {
#pragma unroll
        for (int dx = 0; dx < 3; ++dx) {
          acc[dy * 3 + dx] = fmaf(a, xbc[rows[dy] + cols[dx]], acc[dy * 3 + dx]);
        }
      }
    }
  }

  // write corr: k = tp*9 + s  (0..62)
  if (px < WW && tp < TT - 1) {
#pragma unroll
    for (int s = 0; s < 9; ++s) corr_s[px * 65 + tp * 9 + s] = acc[s];
  }

  // stage conv weights transposed (coalesced global read: idx = o*63+i)
  for (int idx = tid; idx < NO * NI; idx += 512) {
    const int o = idx / NI;
    const int i = idx - o * NI;
    wT_s[i * 64 + o] = conv_w[(size_t)g * NO * NI + idx];
  }
  if (tid < 64) {
    wT_s[63 * 64 + tid] = 0.f;               // zero-pad K row 63
    bias_s[tid] = conv_b[g * NO + tid];
  }
  __syncthreads();

  // ---- Phase B: WMMA GEMM.  16 waves <-> 16 (M-tile, N-tile) jobs ----
  const int lane = tid & 31;
  const int wv = tid >> 5;                   // 0..15
  const int m_base = (wv & 3) * 16;          // pixel tile
  const int n_base = (wv >> 2) * 16;         // output tile
  const int mrow = m_base + (lane & 15);     // A: M index for this lane
  const int ncol = n_base + (lane & 15);     // B/D: N index for this lane
  const int khalf = (lane >> 4) << 1;        // lanes 16-31 take K = k0+2,k0+3

  v8f d = {};
#pragma unroll
  for (int k0 = 0; k0 < 64; k0 += 4) {
    const int kk = k0 + khalf;
    v2f a;
    a[0] = corr_s[mrow * 65 + kk];
    a[1] = corr_s[mrow * 65 + kk + 1];
    v2f b;
    b[0] = wT_s[kk * 64 + ncol];
    b[1] = wT_s[(kk + 1) * 64 + ncol];
    d = __builtin_amdgcn_wmma_f32_16x16x4_f32(
        /*neg_a=*/false, a, /*neg_b=*/false, b,
        /*c_mod=*/(short)0, d, /*reuse_a=*/false, /*reuse_b=*/false);
  }

  // D layout: VGPR v, lane l -> M = v + 8*(l>=16), N = l%16
  const float bs = bias_s[ncol];
  float* wxp = wx + ((size_t)(n * GG + g) * HWSZ + y * WW) * 64;
  const int pbase = m_base + ((lane >> 4) << 3);
#pragma unroll
  for (int v = 0; v < 8; ++v) {
    const int pxv = pbase + v;
    if (pxv < WW) {
      wxp[(size_t)pxv * 64 + ncol] = d[v] + bs;
    }
  }
}

// -------- Kernel 2: out[o,ch] = sum_t wx[o,t]*x[t,chin] + x[o,chin] --------
// One block per (n, g, 64-pixel chunk).  256 threads.
__global__ __launch_bounds__(256) void corrfuse_k2(
    const float* __restrict__ x, const float* __restrict__ wx,
    float* __restrict__ out)
{
  __shared__ float wx_s[64 * 65];            // [px][ot], stride 65

  const int nchunks = HWSZ / 64;             // 49 (exact)
  const int bid = blockIdx.x;
  const int chunk = bid % nchunks;
  const int g = (bid / nchunks) % GG;
  const int n = bid / (nchunks * GG);
  const int p0 = chunk * 64;
  const int tid = threadIdx.x;

  const float* wxp = wx + ((size_t)(n * GG + g) * HWSZ + p0) * 64;
  for (int idx = tid; idx < 4096; idx += 256) {
    const int pxi = idx >> 6;
    const int ot = idx & 63;
    wx_s[pxi * 65 + ot] = wxp[idx];
  }
  __syncthreads();

  const int px = tid & 63;
  const int sub = tid >> 6;                  // 0..3
  const int p = p0 + px;

  for (int cc = sub; cc < CPG; cc += 4) {
    const int chin = cc * GG + g;            // torch (cpg,G) channel split
    float xv[TT];
#pragma unroll
    for (int t = 0; t < TT; ++t) {
      xv[t] = x[((size_t)(n * TT + t) * CC + chin) * HWSZ + p];
    }
#pragma unroll
    for (int o = 0; o < TT; ++o) {
      float s = xv[o];                       // residual: i_in[o] == o
#pragma unroll
      for (int t = 0; t < TT; ++t) {
        s = fmaf(wx_s[px * 65 + o * TT + t], xv[t], s);
      }
      out[((size_t)(n * TT + o) * CC + (g * CPG + cc)) * HWSZ + p] = s;
    }
  }
}

extern "C" void kernel_launch(void* const* d_in, const int* in_sizes, int n_in,
                              void* d_out, int out_size, void* d_ws, size_t ws_size,
                              hipStream_t stream) {
  const float* x  = (const float*)d_in[0];   // (64, 256, 56, 56)
  const float* cw = (const float*)d_in[1];   // (4, 64, 63)
  const float* cb = (const float*)d_in[2];   // (4, 64)
  float* out = (float*)d_out;                // (64, 256, 56, 56)
  float* wx  = (float*)d_ws;                 // NB*G*HW*64 floats = 25.7 MB

  corrfuse_k1<<<NB * GG * HH, 512, 0, stream>>>(x, cw, cb, wx);
  corrfuse_k2<<<NB * GG * (HWSZ / 64), 256, 0, stream>>>(x, wx, out);
}